// M2_GNN_Regressor_38766374814298
// MI455X (gfx1250) — compile-verified
//
#include <hip/hip_runtime.h>
#include <hip/hip_bf16.h>

typedef float v2f __attribute__((ext_vector_type(2)));
typedef float v8f __attribute__((ext_vector_type(8)));

#define HD 64
constexpr int kN = 100000;   // nodes
constexpr int kE = 1600000;  // edges

// silu via hardware v_rcp_f32 (avoids the correctly-rounded div expansion)
__device__ __forceinline__ float silu_f(float x) {
    return x * __builtin_amdgcn_rcpf(1.0f + __expf(-x));
}

// ---------------- t_feat: constant across batch since t == 0 ----------------
// t_emb = [sin(0)x16, cos(0)x16] = [0x16, 1x16]; silu(0)=0, silu(1)=sigmoid(1)
__global__ void k_tfeat(const float* __restrict__ w_t, const float* __restrict__ b_t,
                        float* __restrict__ t_feat) {
    int j = threadIdx.x;
    if (j < HD) {
        const float s1 = 0.7310585786300049f; // silu(1.0)
        float s = 0.f;
        #pragma unroll
        for (int k = 16; k < 32; ++k) s += w_t[k * HD + j];
        t_feat[j] = b_t[j] + s1 * s;
    }
}

__global__ void k_zero(float* __restrict__ p, int n) {
    int i = blockIdx.x * blockDim.x + threadIdx.x;
    if (i < n) p[i] = 0.f;
}

// ---------------- edge MLP: ef = silu(ea @ W1 + b1) @ W2 + b2 ----------------
// One wave per 16-edge tile; layer1 (K=2) in VALU producing A-frags, layer2 WMMA.
__global__ void k_edge_mlp(const float* __restrict__ edge_attr,
                           const float* __restrict__ w1, const float* __restrict__ b1,
                           const float* __restrict__ w2, const float* __restrict__ b2,
                           float* __restrict__ ef) {
    __shared__ float sW2[HD * HD];
    __shared__ float sW1[2 * HD];
    __shared__ float sB1[HD];
    __shared__ float sB2[HD];
    for (int i = threadIdx.x; i < HD * HD; i += blockDim.x) sW2[i] = w2[i];
    for (int i = threadIdx.x; i < 2 * HD; i += blockDim.x)  sW1[i] = w1[i];
    for (int i = threadIdx.x; i < HD; i += blockDim.x) { sB1[i] = b1[i]; sB2[i] = b2[i]; }
    __syncthreads();

    const int wave = threadIdx.x >> 5;
    const int lane = threadIdx.x & 31;
    const int row  = lane & 15;   // M for A-frags, N for B/D frags
    const int hi   = lane >> 4;
    const int kb   = hi * 2;      // K sub-offset per ISA 16x4 A layout
    const int e0   = (blockIdx.x * 8 + wave) * 16;
    if (e0 >= kE) return;

    const float ea0 = edge_attr[(size_t)(e0 + row) * 2 + 0];
    const float ea1 = edge_attr[(size_t)(e0 + row) * 2 + 1];

    v2f a[16];
    #pragma unroll
    for (int kk = 0; kk < 16; ++kk) {
        int c0 = kk * 4 + kb;
        a[kk].x = silu_f(ea0 * sW1[c0]     + ea1 * sW1[HD + c0]     + sB1[c0]);
        a[kk].y = silu_f(ea0 * sW1[c0 + 1] + ea1 * sW1[HD + c0 + 1] + sB1[c0 + 1]);
    }

    #pragma unroll
    for (int nt = 0; nt < 4; ++nt) {
        const int nc = nt * 16 + row;
        v8f acc;
        #pragma unroll
        for (int v = 0; v < 8; ++v) acc[v] = sB2[nc];  // bias folded into C
        #pragma unroll
        for (int kk = 0; kk < 16; ++kk) {
            int c0 = kk * 4 + kb;
            v2f b;
            b.x = sW2[c0 * HD + nc];
            b.y = sW2[(c0 + 1) * HD + nc];
            acc = __builtin_amdgcn_wmma_f32_16x16x4_f32(false, a[kk], false, b,
                                                        (short)0, acc, false, false);
        }
        #pragma unroll
        for (int v = 0; v < 8; ++v)
            ef[(size_t)(e0 + v + hi * 8) * HD + nc] = acc[v];
    }
}

// ---------------- node MLP with t_feat folded into layer-2 bias ----------------
__global__ void k_node_mlp(const float* __restrict__ cond,
                           const float* __restrict__ w1, const float* __restrict__ b1,
                           const float* __restrict__ w2, const float* __restrict__ b2,
                           const float* __restrict__ t_feat,
                           float* __restrict__ h) {
    __shared__ float sW2[HD * HD];
    __shared__ float sW1[7 * HD];
    __shared__ float sB1[HD];
    __shared__ float sB2[HD];
    for (int i = threadIdx.x; i < HD * HD; i += blockDim.x) sW2[i] = w2[i];
    for (int i = threadIdx.x; i < 7 * HD; i += blockDim.x)  sW1[i] = w1[i];
    for (int i = threadIdx.x; i < HD; i += blockDim.x) {
        sB1[i] = b1[i];
        sB2[i] = b2[i] + t_feat[i];  // h += t_feat[batch] is node-independent
    }
    __syncthreads();

    const int wave = threadIdx.x >> 5;
    const int lane = threadIdx.x & 31;
    const int row  = lane & 15;
    const int hi   = lane >> 4;
    const int kb   = hi * 2;
    const int n0   = (blockIdx.x * 8 + wave) * 16;
    if (n0 >= kN) return;

    float cv[6];
    const float* cr = cond + (size_t)(n0 + row) * 6;
    #pragma unroll
    for (int k = 0; k < 6; ++k) cv[k] = cr[k];

    v2f a[16];
    #pragma unroll
    for (int kk = 0; kk < 16; ++kk) {
        int c0 = kk * 4 + kb;
        float z0 = sB1[c0], z1 = sB1[c0 + 1];       // x_in col 0 is zero
        #pragma unroll
        for (int k = 0; k < 6; ++k) {
            z0 += cv[k] * sW1[(k + 1) * HD + c0];
            z1 += cv[k] * sW1[(k + 1) * HD + c0 + 1];
        }
        a[kk].x = silu_f(z0);
        a[kk].y = silu_f(z1);
    }

    #pragma unroll
    for (int nt = 0; nt < 4; ++nt) {
        const int nc = nt * 16 + row;
        v8f acc;
        #pragma unroll
        for (int v = 0; v < 8; ++v) acc[v] = sB2[nc];
        #pragma unroll
        for (int kk = 0; kk < 16; ++kk) {
            int c0 = kk * 4 + kb;
            v2f b;
            b.x = sW2[c0 * HD + nc];
            b.y = sW2[(c0 + 1) * HD + nc];
            acc = __builtin_amdgcn_wmma_f32_16x16x4_f32(false, a[kk], false, b,
                                                        (short)0, acc, false, false);
        }
        #pragma unroll
        for (int v = 0; v < 8; ++v)
            h[(size_t)(n0 + v + hi * 8) * HD + nc] = acc[v];
    }
}

// ---------------- message + segment-sum: agg[dst] += relu(h[src] + ef) ----------------
// One wave per edge (wave-uniform edge id -> scalar index loads),
// 2 columns per lane, fp32 atomics (hot in the 192MB L2).
__global__ void k_scatter(const float* __restrict__ h, const float* __restrict__ ef,
                          const int* __restrict__ eidx, float* __restrict__ agg) {
    const int e = blockIdx.x * 8 + (threadIdx.x >> 5);   // uniform across the wave
    if (e >= kE) return;
    const int lane = threadIdx.x & 31;
    const int src = eidx[e];          // scalar loads (uniform address)
    const int dst = eidx[kE + e];
    const int c = lane * 2;
    const float2 hv = *(const float2*)(h  + (size_t)src * HD + c);
    const float2 ev = *(const float2*)(ef + (size_t)e   * HD + c);
    const float m0 = fmaxf(hv.x + ev.x, 0.f);
    const float m1 = fmaxf(hv.y + ev.y, 0.f);
    atomicAdd(&agg[(size_t)dst * HD + c],     m0);
    atomicAdd(&agg[(size_t)dst * HD + c + 1], m1);
}

// ---------------- GINE dense part: h = silu(mlp2(h + agg)) (in place) ----------------
__global__ void k_conv_dense(float* __restrict__ h, const float* __restrict__ agg,
                             const float* __restrict__ w1, const float* __restrict__ b1,
                             const float* __restrict__ w2, const float* __restrict__ b2) {
    __shared__ float sW1[HD * HD];
    __shared__ float sW2[HD * HD];
    __shared__ float sB1[HD];
    __shared__ float sB2[HD];
    __shared__ float sX[8][16 * HD];   // per-wave staging, reused x -> z
    for (int i = threadIdx.x; i < HD * HD; i += blockDim.x) { sW1[i] = w1[i]; sW2[i] = w2[i]; }
    for (int i = threadIdx.x; i < HD; i += blockDim.x) { sB1[i] = b1[i]; sB2[i] = b2[i]; }
    __syncthreads();

    const int wave = threadIdx.x >> 5;
    const int lane = threadIdx.x & 31;
    const int row  = lane & 15;
    const int hi   = lane >> 4;
    const int kb   = hi * 2;
    const int n0   = (blockIdx.x * 8 + wave) * 16;
    if (n0 >= kN) return;

    float* X = &sX[wave][0];
    const size_t base = (size_t)n0 * HD;
    #pragma unroll
    for (int i = 0; i < 16; ++i) {          // coalesced: x = h + agg
        int off = i * HD + lane * 2;
        X[off]     = h[base + off]     + agg[base + off];
        X[off + 1] = h[base + off + 1] + agg[base + off + 1];
    }

    v2f a[16];
    #pragma unroll
    for (int kk = 0; kk < 16; ++kk) {
        int c0 = kk * 4 + kb;
        a[kk].x = X[row * HD + c0];
        a[kk].y = X[row * HD + c0 + 1];
    }

    // layer 1 -> silu -> back into X (A-frag friendly row-major layout)
    #pragma unroll
    for (int nt = 0; nt < 4; ++nt) {
        const int nc = nt * 16 + row;
        v8f acc;
        #pragma unroll
        for (int v = 0; v < 8; ++v) acc[v] = sB1[nc];
        #pragma unroll
        for (int kk = 0; kk < 16; ++kk) {
            int c0 = kk * 4 + kb;
            v2f b;
            b.x = sW1[c0 * HD + nc];
            b.y = sW1[(c0 + 1) * HD + nc];
            acc = __builtin_amdgcn_wmma_f32_16x16x4_f32(false, a[kk], false, b,
                                                        (short)0, acc, false, false);
        }
        #pragma unroll
        for (int v = 0; v < 8; ++v)
            X[(v + hi * 8) * HD + nc] = silu_f(acc[v]);
    }

    v2f a2[16];
    #pragma unroll
    for (int kk = 0; kk < 16; ++kk) {
        int c0 = kk * 4 + kb;
        a2[kk].x = X[row * HD + c0];
        a2[kk].y = X[row * HD + c0 + 1];
    }

    // layer 2 -> outer silu -> h in place
    #pragma unroll
    for (int nt = 0; nt < 4; ++nt) {
        const int nc = nt * 16 + row;
        v8f acc;
        #pragma unroll
        for (int v = 0; v < 8; ++v) acc[v] = sB2[nc];
        #pragma unroll
        for (int kk = 0; kk < 16; ++kk) {
            int c0 = kk * 4 + kb;
            v2f b;
            b.x = sW2[c0 * HD + nc];
            b.y = sW2[(c0 + 1) * HD + nc];
            acc = __builtin_amdgcn_wmma_f32_16x16x4_f32(false, a2[kk], false, b,
                                                        (short)0, acc, false, false);
        }
        #pragma unroll
        for (int v = 0; v < 8; ++v)
            h[(size_t)(n0 + v + hi * 8) * HD + nc] = silu_f(acc[v]);
    }
}

// ---------------- final MLP: out = silu(h @ Wf1 + bf1) @ wf2 + bf2 ----------------
__global__ void k_final(const float* __restrict__ h,
                        const float* __restrict__ w1, const float* __restrict__ b1,
                        const float* __restrict__ w2, const float* __restrict__ b2,
                        float* __restrict__ out) {
    __shared__ float sW1[HD * HD];
    __shared__ float sB1[HD];
    __shared__ float sV[HD];
    __shared__ float sX[8][16 * HD];
    for (int i = threadIdx.x; i < HD * HD; i += blockDim.x) sW1[i] = w1[i];
    for (int i = threadIdx.x; i < HD; i += blockDim.x) { sB1[i] = b1[i]; sV[i] = w2[i]; }
    __syncthreads();

    const int wave = threadIdx.x >> 5;
    const int lane = threadIdx.x & 31;
    const int row  = lane & 15;
    const int hi   = lane >> 4;
    const int kb   = hi * 2;
    const int n0   = (blockIdx.x * 8 + wave) * 16;
    if (n0 >= kN) return;

    float* X = &sX[wave][0];
    const size_t base = (size_t)n0 * HD;
    #pragma unroll
    for (int i = 0; i < 16; ++i) {
        int off = i * HD + lane * 2;
        X[off]     = h[base + off];
        X[off + 1] = h[base + off + 1];
    }

    v2f a[16];
    #pragma unroll
    for (int kk = 0; kk < 16; ++kk) {
        int c0 = kk * 4 + kb;
        a[kk].x = X[row * HD + c0];
        a[kk].y = X[row * HD + c0 + 1];
    }

    #pragma unroll
    for (int nt = 0; nt < 4; ++nt) {
        const int nc = nt * 16 + row;
        v8f acc;
        #pragma unroll
        for (int v = 0; v < 8; ++v) acc[v] = sB1[nc];
        #pragma unroll
        for (int kk = 0; kk < 16; ++kk) {
            int c0 = kk * 4 + kb;
            v2f b;
            b.x = sW1[c0 * HD + nc];
            b.y = sW1[(c0 + 1) * HD + nc];
            acc = __builtin_amdgcn_wmma_f32_16x16x4_f32(false, a[kk], false, b,
                                                        (short)0, acc, false, false);
        }
        #pragma unroll
        for (int v = 0; v < 8; ++v)
            X[(v + hi * 8) * HD + nc] = silu_f(acc[v]);
    }

    if (lane < 16) {  // per-node dot with wf2
        float accv = b2[0];
        #pragma unroll
        for (int c = 0; c < HD; ++c) accv += X[lane * HD + c] * sV[c];
        out[n0 + lane] = accv;
    }
}

extern "C" void kernel_launch(void* const* d_in, const int* in_sizes, int n_in,
                              void* d_out, int out_size, void* d_ws, size_t ws_size,
                              hipStream_t stream) {
    const float* condition  = (const float*)d_in[0];
    const float* edge_attr  = (const float*)d_in[1];
    const int*   edge_index = (const int*)  d_in[2];
    // d_in[3] = batch (unused: t_feat is batch-independent since t == 0)
    const float* w_n1 = (const float*)d_in[4];  const float* b_n1 = (const float*)d_in[5];
    const float* w_n2 = (const float*)d_in[6];  const float* b_n2 = (const float*)d_in[7];
    const float* w_e1 = (const float*)d_in[8];  const float* b_e1 = (const float*)d_in[9];
    const float* w_e2 = (const float*)d_in[10]; const float* b_e2 = (const float*)d_in[11];
    const float* w_t  = (const float*)d_in[12]; const float* b_t  = (const float*)d_in[13];
    const float* wca[3] = {(const float*)d_in[14], (const float*)d_in[18], (const float*)d_in[22]};
    const float* bca[3] = {(const float*)d_in[15], (const float*)d_in[19], (const float*)d_in[23]};
    const float* wcb[3] = {(const float*)d_in[16], (const float*)d_in[20], (const float*)d_in[24]};
    const float* bcb[3] = {(const float*)d_in[17], (const float*)d_in[21], (const float*)d_in[25]};
    const float* w_f1 = (const float*)d_in[26]; const float* b_f1 = (const float*)d_in[27];
    const float* w_f2 = (const float*)d_in[28]; const float* b_f2 = (const float*)d_in[29];

    char* ws = (char*)d_ws;
    float* t_feat = (float*)ws;                                       // 64 floats
    float* h      = (float*)(ws + 256);                               // N*64
    float* agg    = (float*)(ws + 256 + (size_t)kN * HD * 4);         // N*64
    float* ef     = (float*)(ws + 256 + 2 * (size_t)kN * HD * 4);     // E*64

    const int nodeTiles  = (kN + 15) / 16;          // 6250
    const int nodeBlocks = (nodeTiles + 7) / 8;     // 782
    const int edgeBlocks = kE / (16 * 8);           // 12500 (exact)

    k_tfeat<<<1, 64, 0, stream>>>(w_t, b_t, t_feat);
    k_edge_mlp<<<edgeBlocks, 256, 0, stream>>>(edge_attr, w_e1, b_e1, w_e2, b_e2, ef);
    k_node_mlp<<<nodeBlocks, 256, 0, stream>>>(condition, w_n1, b_n1, w_n2, b_n2, t_feat, h);

    for (int c = 0; c < 3; ++c) {
        k_zero<<<(kN * HD + 255) / 256, 256, 0, stream>>>(agg, kN * HD);
        k_scatter<<<kE / 8, 256, 0, stream>>>(h, ef, edge_index, agg);
        k_conv_dense<<<nodeBlocks, 256, 0, stream>>>(h, agg, wca[c], bca[c], wcb[c], bcb[c]);
    }
    k_final<<<nodeBlocks, 256, 0, stream>>>(h, w_f1, b_f1, w_f2, b_f2, (float*)d_out);
}